// MeanFusion_41661182771841
// MI455X (gfx1250) — compile-verified
//
#include <hip/hip_runtime.h>

// MeanFusion warp+ragged-mean for MI455X (gfx1250).
// x:    (32, 64, 200, 256) f32
// rec:  (8,) i32   (sums to 32)
// tmat: (8, 5, 5, 4, 4) f32
// out:  (8, 64, 200, 256) f32
//
// Memory-bound gather/warp. Per-(h,w) coordinate math is done once per agent
// and amortized over a 16-channel chunk; the inner loop is a batched clause of
// global_load_b32, 4 independent FMA accumulators, one coalesced store, and a
// global_prefetch_b8 pair for the next channel plane.

namespace {

constexpr int B = 8, L = 5, C = 64, H = 200, W = 256;
constexpr int HW = H * W;
constexpr int CCHUNK = 16;                 // channels per block
constexpr float DISCRETE_RATIO = 0.4f;
constexpr float DOWNSAMPLE_RATE = 2.0f;

struct AgentTaps {
  const float* p00;
  const float* p01;
  const float* p10;
  const float* p11;
  float w00, w01, w10, w11;
};

template <int NA>
__device__ __forceinline__ void channel_loop(const AgentTaps* __restrict__ t,
                                             float* __restrict__ outp,
                                             float inv_len) {
  size_t off = 0;
  #pragma unroll 2
  for (int c = 0; c < CCHUNK; ++c) {
    // Batch all taps of this channel into temporaries first so the compiler
    // emits one load clause, then consumes with few loadcnt waits.
    float v00[NA], v01[NA], v10[NA], v11[NA];
    #pragma unroll
    for (int a = 0; a < NA; ++a) {
      v00[a] = t[a].p00[off];
      v01[a] = t[a].p01[off];
      v10[a] = t[a].p10[off];
      v11[a] = t[a].p11[off];
    }
    off += HW;
    if (c + 1 < CCHUNK) {
      // gfx1250 global_prefetch_b8: pull next channel plane's tap rows early.
      __builtin_prefetch(t[0].p00 + off, 0, 0);
      __builtin_prefetch(t[0].p10 + off, 0, 0);
    }
    // Four independent accumulator chains (one per tap position).
    float acc0 = 0.0f, acc1 = 0.0f, acc2 = 0.0f, acc3 = 0.0f;
    #pragma unroll
    for (int a = 0; a < NA; ++a) {
      acc0 = fmaf(v00[a], t[a].w00, acc0);
      acc1 = fmaf(v01[a], t[a].w01, acc1);
      acc2 = fmaf(v10[a], t[a].w10, acc2);
      acc3 = fmaf(v11[a], t[a].w11, acc3);
    }
    outp[(size_t)c * HW] = ((acc0 + acc1) + (acc2 + acc3)) * inv_len;
  }
}

__global__ __launch_bounds__(W) void mean_fusion_warp_kernel(
    const float* __restrict__ x, const int* __restrict__ record_len,
    const float* __restrict__ tmat, float* __restrict__ out) {
  const int w    = threadIdx.x;   // 0..255  (W == 256)
  const int h    = blockIdx.x;    // 0..199
  const int cblk = blockIdx.y;    // 0..3
  const int b    = blockIdx.z;    // 0..7

  // Ragged offsets: start = sum(record_len[0..b-1]), len = record_len[b].
  int start = 0, len = 0;
  #pragma unroll
  for (int i = 0; i < B; ++i) {
    int r = record_len[i];
    if (i < b) start += r;
    if (i == b) len = r;
  }
  len   = __builtin_amdgcn_readfirstlane(len);    // uniform -> scalar branch
  start = __builtin_amdgcn_readfirstlane(start);
  const int na = (len > L) ? L : (len < 1 ? 1 : len);

  // affine_grid base coords (align_corners=False)
  const float gx = ((float)w + 0.5f) * (2.0f / (float)W) - 1.0f;
  const float gy = ((float)h + 0.5f) * (2.0f / (float)H) - 1.0f;

  const float sxs = 2.0f / (DOWNSAMPLE_RATE * DISCRETE_RATIO * (float)W);
  const float sys = 2.0f / (DOWNSAMPLE_RATE * DISCRETE_RATIO * (float)H);

  AgentTaps taps[L];
  for (int a = 0; a < na; ++a) {
    // theta = t[b, 0, a][:2, [0,1,3]] with the reference's scalings
    const float* tb = tmat + ((size_t)(b * L + 0) * L + a) * 16;
    const float th00 = tb[0];
    const float th01 = tb[1] * ((float)H / (float)W);
    const float th02 = tb[3] * sxs;
    const float th10 = tb[4] * ((float)W / (float)H);
    const float th11 = tb[5];
    const float th12 = tb[7] * sys;

    // grid coords -> source pixel coords
    const float gsx = fmaf(th00, gx, fmaf(th01, gy, th02));
    const float gsy = fmaf(th10, gx, fmaf(th11, gy, th12));
    const float fx = ((gsx + 1.0f) * (float)W - 1.0f) * 0.5f;
    const float fy = ((gsy + 1.0f) * (float)H - 1.0f) * 0.5f;

    const float x0f = floorf(fx);
    const float y0f = floorf(fy);
    const float wx1 = fx - x0f;
    const float wy1 = fy - y0f;
    const float wx0 = 1.0f - wx1;
    const float wy0 = 1.0f - wy1;

    const int ix0 = (int)x0f;
    const int iy0 = (int)y0f;
    const int ix1 = ix0 + 1;
    const int iy1 = iy0 + 1;

    const bool vx0 = (ix0 >= 0) && (ix0 < W);
    const bool vx1 = (ix1 >= 0) && (ix1 < W);
    const bool vy0 = (iy0 >= 0) && (iy0 < H);
    const bool vy1 = (iy1 >= 0) && (iy1 < H);

    const int cx0 = min(max(ix0, 0), W - 1);
    const int cx1 = min(max(ix1, 0), W - 1);
    const int cy0 = min(max(iy0, 0), H - 1);
    const int cy1 = min(max(iy1, 0), H - 1);

    const float* base =
        x + ((size_t)(start + a) * C + (size_t)cblk * CCHUNK) * HW;
    taps[a].p00 = base + cy0 * W + cx0;
    taps[a].p01 = base + cy0 * W + cx1;
    taps[a].p10 = base + cy1 * W + cx0;
    taps[a].p11 = base + cy1 * W + cx1;
    // zeros-padding: fold validity into the bilinear weight
    taps[a].w00 = wx0 * wy0 * ((vx0 && vy0) ? 1.0f : 0.0f);
    taps[a].w01 = wx1 * wy0 * ((vx1 && vy0) ? 1.0f : 0.0f);
    taps[a].w10 = wx0 * wy1 * ((vx0 && vy1) ? 1.0f : 0.0f);
    taps[a].w11 = wx1 * wy1 * ((vx1 && vy1) ? 1.0f : 0.0f);
  }

  float* outp =
      out + (((size_t)b * C + (size_t)cblk * CCHUNK) * H + h) * W + w;
  const float inv_len = 1.0f / (float)len;

  switch (na) {
    case 1: channel_loop<1>(taps, outp, inv_len); break;
    case 2: channel_loop<2>(taps, outp, inv_len); break;
    case 3: channel_loop<3>(taps, outp, inv_len); break;
    case 4: channel_loop<4>(taps, outp, inv_len); break;
    default: channel_loop<5>(taps, outp, inv_len); break;
  }
}

}  // namespace

extern "C" void kernel_launch(void* const* d_in, const int* in_sizes, int n_in,
                              void* d_out, int out_size, void* d_ws,
                              size_t ws_size, hipStream_t stream) {
  (void)in_sizes; (void)n_in; (void)out_size; (void)d_ws; (void)ws_size;
  const float* x          = (const float*)d_in[0];
  const int*   record_len = (const int*)d_in[1];
  const float* tmat       = (const float*)d_in[2];
  float*       out        = (float*)d_out;

  dim3 grid(H, C / CCHUNK, B);   // (200, 4, 8)
  dim3 block(W);                 // 256 threads = 8 wave32
  hipLaunchKernelGGL(mean_fusion_warp_kernel, grid, block, 0, stream,
                     x, record_len, tmat, out);
}